// ParticleCloud_41008347742440
// MI455X (gfx1250) — compile-verified
//
#include <hip/hip_runtime.h>
#include <hip/hip_bf16.h>
#include <math.h>

// ---------------------------------------------------------------------------
// ParticleCloud: per-batch kNN (K=3) -> edge MLP (12->32->32, relu) ->
// mean over K,N -> 32->2 head -> softmax.
// B=256, N=512, F=6, K=3.
//
// MI455X reasoning: total tensor data ~3MB (L2 is 192MB) => everything is
// cache resident; HBM is irrelevant. Dominant work is the O(B*N^2)=67M pair
// kNN (VALU, LDS-resident coords). The edge MLP (~1.1 GFLOP, fp32 needed for
// fidelity) is mapped onto v_wmma_f32_16x16x4_f32: 16 edges per A tile,
// fp32 accumulate. One workgroup per batch => pooled reduction stays in LDS.
// ---------------------------------------------------------------------------

typedef __attribute__((ext_vector_type(2))) float v2f;
typedef __attribute__((ext_vector_type(8))) float v8f;

#define BATCH 256
#define NPTS  512
#define FEAT  6
#define KNN   3
#define EPB   (NPTS * KNN)      // 1536 edges per batch
#define TPB   (EPB / 16)        // 96 16-edge tiles per batch
#define NWAVE 8                 // 256 threads / wave32

// ---------------------------------------------------------------------------
// Kernel 1: kNN. One block per batch, one thread per point. Coords staged in
// LDS (4KB); each thread scans all 512 points keeping a sorted top-3
// (strict '<' => ties keep the lower index, matching jax.lax.top_k).
// ---------------------------------------------------------------------------
__global__ __launch_bounds__(NPTS) void knn_kernel(const float* __restrict__ x,
                                                   int* __restrict__ idx_out) {
    __shared__ float cx[NPTS];
    __shared__ float cy[NPTS];
    const int b = blockIdx.x;
    const int i = threadIdx.x;
    const float* xb = x + (size_t)b * NPTS * FEAT;
    cx[i] = xb[i * FEAT + 1];
    cy[i] = xb[i * FEAT + 2];
    __syncthreads();

    const float px = cx[i], py = cy[i];
    float d0 = 3.0e38f, d1 = 3.0e38f, d2 = 3.0e38f;
    int   i0 = 0, i1 = 0, i2 = 0;
    for (int j = 0; j < NPTS; ++j) {
        float dx = px - cx[j];
        float dy = py - cy[j];
        float d  = dx * dx + dy * dy;
        if (j == i) d += 1.0e9f;          // matches reference's +1e9 diagonal
        if (d < d2) {
            if (d < d1) {
                if (d < d0) { d2 = d1; i2 = i1; d1 = d0; i1 = i0; d0 = d; i0 = j; }
                else        { d2 = d1; i2 = i1; d1 = d;  i1 = j; }
            } else          { d2 = d;  i2 = j; }
        }
    }
    int* o = idx_out + ((size_t)b * NPTS + i) * KNN;
    o[0] = i0; o[1] = i1; o[2] = i2;
}

// ---------------------------------------------------------------------------
// Kernel 2: edge MLP via v_wmma_f32_16x16x4_f32, pooled sum per batch.
// One block (8 waves) per batch; each wave owns 12 tiles of 16 edges.
//
// WMMA fp32 lane layouts (ISA 7.12.2):
//   A 16x4 : lanes 0-15 -> M=lane, VGPR{0,1}=K{0,1}; lanes 16-31 -> K{2,3}
//   B 4x16 : VGPR r, lanes 0-15 -> K=r, N=lane; lanes 16-31 -> K=r+2
//   C 16x16: VGPR r, lanes 0-15 -> M=r, N=lane; lanes 16-31 -> M=r+8
// ---------------------------------------------------------------------------
__global__ __launch_bounds__(256) void edge_mlp_kernel(
    const float* __restrict__ x, const int* __restrict__ idx,
    const float* __restrict__ W1, const float* __restrict__ b1,
    const float* __restrict__ W2, const float* __restrict__ b2,
    float* __restrict__ pooled) {
    __shared__ float xs[NPTS * FEAT];        // 12 KB: this batch's features
    __shared__ int   nbr[EPB];               // 6 KB: neighbor indices
    __shared__ float w1s[12 * 32];           // 1.5 KB
    __shared__ float w2s[32 * 32];           // 4 KB
    __shared__ float b1s[32], b2s[32];
    __shared__ float h1s[NWAVE][16 * 32];    // 16 KB: per-wave layer-1 output
    __shared__ float pool[32];

    const int b    = blockIdx.x;
    const int tid  = threadIdx.x;
    const int lane = tid & 31;
    const int wave = tid >> 5;

    for (int t = tid; t < NPTS * FEAT; t += 256) xs[t]  = x[(size_t)b * NPTS * FEAT + t];
    for (int t = tid; t < EPB;        t += 256) nbr[t] = idx[(size_t)b * EPB + t];
    for (int t = tid; t < 12 * 32;    t += 256) w1s[t] = W1[t];
    for (int t = tid; t < 32 * 32;    t += 256) w2s[t] = W2[t];
    if (tid < 32) { b1s[tid] = b1[tid]; b2s[tid] = b2[tid]; pool[tid] = 0.0f; }
    __syncthreads();

    const int m  = lane & 15;   // A-row / D-col within the lane half
    const int hi = lane >> 4;   // 0: lanes 0-15, 1: lanes 16-31
    float* myh1 = h1s[wave];

    for (int t = wave; t < TPB; t += NWAVE) {   // uniform trip count: EXEC stays full
        const int e  = t * 16 + m;              // edge id within batch
        const int n  = e / 3;                   // source node
        const int kk = e - n * 3;               // neighbor slot
        const int j  = nbr[n * 3 + kk];
        const float* xn = &xs[n * FEAT];
        const float* xj = &xs[j * FEAT];

        // ----- layer 1: edge(16x12) @ W1(12x32), two 16-col halves, 3 K-steps
        for (int h = 0; h < 2; ++h) {
            v8f acc = {};
#pragma unroll
            for (int s = 0; s < 3; ++s) {
                const int kb = 4 * s + 2 * hi;
                v2f a, bm;
#pragma unroll
                for (int c = 0; c < 2; ++c) {
                    const int f = kb + c;       // edge feature = [xi, xj-xi]
                    a[c]  = (f < 6) ? xn[f] : (xj[f - 6] - xn[f - 6]);
                    bm[c] = w1s[(kb + c) * 32 + 16 * h + m];
                }
                acc = __builtin_amdgcn_wmma_f32_16x16x4_f32(
                    false, a, false, bm, (short)0, acc, false, false);
            }
            const float bias = b1s[16 * h + m];
#pragma unroll
            for (int r = 0; r < 8; ++r) {
                float v = acc[r] + bias;
                v = v > 0.0f ? v : 0.0f;
                myh1[(r + 8 * hi) * 32 + 16 * h + m] = v;   // D-layout -> row-major LDS
            }
        }
        // LDS ops from one wave are processed in-order (ISA 5.7.3): the loads
        // below observe the stores above without a barrier.

        // ----- layer 2: h1(16x32) @ W2(32x32), two halves, 8 K-steps
        for (int h = 0; h < 2; ++h) {
            v8f acc = {};
#pragma unroll
            for (int s = 0; s < 8; ++s) {
                const int kb = 4 * s + 2 * hi;
                v2f a, bm;
#pragma unroll
                for (int c = 0; c < 2; ++c) {
                    a[c]  = myh1[m * 32 + kb + c];
                    bm[c] = w2s[(kb + c) * 32 + 16 * h + m];
                }
                acc = __builtin_amdgcn_wmma_f32_16x16x4_f32(
                    false, a, false, bm, (short)0, acc, false, false);
            }
            const float bias = b2s[16 * h + m];
            float s8 = 0.0f;
#pragma unroll
            for (int r = 0; r < 8; ++r) {
                float v = acc[r] + bias;
                v = v > 0.0f ? v : 0.0f;
                s8 += v;                        // sum over this lane's 8 rows
            }
            atomicAdd(&pool[16 * h + m], s8);   // ds_add_f32
        }
    }
    __syncthreads();
    // mean over K then N == sum over all N*K edges / (N*K)
    if (tid < 32) pooled[b * 32 + tid] = pool[tid] * (1.0f / (float)EPB);
}

// ---------------------------------------------------------------------------
// Kernel 3: 32->2 head + softmax. One thread per batch.
// ---------------------------------------------------------------------------
__global__ __launch_bounds__(256) void head_kernel(const float* __restrict__ pooled,
                                                   const float* __restrict__ Wo,
                                                   const float* __restrict__ bo,
                                                   float* __restrict__ out) {
    const int b = blockIdx.x * blockDim.x + threadIdx.x;
    if (b >= BATCH) return;
    float l0 = bo[0], l1 = bo[1];
#pragma unroll
    for (int c = 0; c < 32; ++c) {
        const float p = pooled[b * 32 + c];
        l0 += p * Wo[c * 2 + 0];
        l1 += p * Wo[c * 2 + 1];
    }
    const float mx = fmaxf(l0, l1);
    const float e0 = expf(l0 - mx);
    const float e1 = expf(l1 - mx);
    const float inv = 1.0f / (e0 + e1);
    out[b * 2 + 0] = e0 * inv;
    out[b * 2 + 1] = e1 * inv;
}

// ---------------------------------------------------------------------------
extern "C" void kernel_launch(void* const* d_in, const int* in_sizes, int n_in,
                              void* d_out, int out_size, void* d_ws, size_t ws_size,
                              hipStream_t stream) {
    (void)in_sizes; (void)n_in; (void)out_size; (void)ws_size;
    const float* x  = (const float*)d_in[0];
    const float* W1 = (const float*)d_in[1];
    const float* b1 = (const float*)d_in[2];
    const float* W2 = (const float*)d_in[3];
    const float* b2 = (const float*)d_in[4];
    const float* Wo = (const float*)d_in[5];
    const float* bo = (const float*)d_in[6];
    float* out = (float*)d_out;

    int*   idx    = (int*)d_ws;                                              // B*N*K ints
    float* pooled = (float*)((char*)d_ws + (size_t)BATCH * NPTS * KNN * 4);  // B*32 floats

    knn_kernel<<<BATCH, NPTS, 0, stream>>>(x, idx);
    edge_mlp_kernel<<<BATCH, 256, 0, stream>>>(x, idx, W1, b1, W2, b2, pooled);
    head_kernel<<<1, 256, 0, stream>>>(pooled, Wo, bo, out);
}